// StickBreaking_65953517797987
// MI455X (gfx1250) — compile-verified
//
#include <hip/hip_runtime.h>
#include <hip/hip_bf16.h>
#include <cstdint>

// Stick-breaking recurrence, B=32 batches of N=512 x 512.
// One workgroup per batch; thread t owns column t.
// Row phases:  A) load row (async-prefetched), sigmoid, fut = max(0, mask - cs)
//              B) suffix sums of fut via V_WMMA_F32_16X16X4_F32 (chunked) + carries
//              C) serial row_sum scan by one lane over LDS-packed float4
//              D) p out, column sums update.

static constexpr int kN  = 512;   // matrix dim
static constexpr int kNT = 512;   // threads per block (16 wave32s)

typedef float v2f __attribute__((ext_vector_type(2)));
typedef float v8f __attribute__((ext_vector_type(8)));

#if defined(__HIP_DEVICE_COMPILE__) && defined(__gfx1250__)
  #define CDNA5_DEV 1
#endif

#if defined(CDNA5_DEV) && __has_builtin(__builtin_amdgcn_wmma_f32_16x16x4_f32)
  #define USE_WMMA 1
#endif

#if defined(CDNA5_DEV)
  #define USE_ASYNC 1
#endif

__device__ __forceinline__ unsigned lds_off_of(const void* p) {
  // HW uses addr[31:0] of a generic LDS address as the LDS offset.
  return (unsigned)(unsigned long long)(uintptr_t)p;
}

#if defined(USE_ASYNC)
__device__ __forceinline__ void async_load_b128(unsigned lds_addr, const void* gaddr) {
  asm volatile("global_load_async_to_lds_b128 %0, %1, off"
               :: "v"(lds_addr), "v"((unsigned long long)(uintptr_t)gaddr)
               : "memory");
}
__device__ __forceinline__ void wait_asynccnt0() {
#if __has_builtin(__builtin_amdgcn_s_wait_asynccnt)
  __builtin_amdgcn_s_wait_asynccnt(0);
#else
  asm volatile("s_wait_asynccnt 0" ::: "memory");
#endif
}
#endif

__global__ __launch_bounds__(kNT)
void StickBreaking_65953517797987_kernel(const float* __restrict__ x,
                                         const float* __restrict__ xmask,
                                         float* __restrict__ out) {
  __shared__ float  s_fut[kN];        // fut per column
  __shared__ float  s_suf[kN];        // within-chunk (or global) inclusive suffix sums
  __shared__ float  s_carry[32];      // cross-chunk exclusive-suffix carries
  __shared__ float4 s_pack[kN];       // {sigmoid(b), mask, mfm, cs} per column
  __shared__ float  s_p[kN];          // row result
  __shared__ float  s_xbuf[2][kN];    // double-buffered logits row
  __shared__ float  s_mbuf[2][kN];    // double-buffered mask row

  const int    tid  = threadIdx.x;
  const int    bat  = blockIdx.x;
  const size_t base = (size_t)bat * kN * kN;

  float cs = 0.0f;  // column partial sum for column `tid` (rows < m)

#if !defined(USE_WMMA)
  if (tid < 32) s_carry[tid] = 0.0f;
#endif

#if defined(USE_ASYNC)
  // prefetch row 0 (128 lanes x b128 for x, 128 lanes for mask; separate waves)
  if (tid < 128) {
    async_load_b128(lds_off_of(&s_xbuf[0][tid * 4]), x + base + (size_t)tid * 4);
  } else if (tid >= 256 && tid < 384) {
    const int t = tid - 256;
    async_load_b128(lds_off_of(&s_mbuf[0][t * 4]), xmask + base + (size_t)t * 4);
  }
#endif

  for (int m = 0; m < kN; ++m) {
    const int nb = m & 1;
    float xv, mv;

#if defined(USE_ASYNC)
    wait_asynccnt0();        // each wave waits on its own async loads
    __syncthreads();         // then everyone sees the full buffers
    xv = s_xbuf[nb][tid];
    mv = s_mbuf[nb][tid];
    if (m + 1 < kN) {        // kick off next row into the other buffer
      if (tid < 128) {
        async_load_b128(lds_off_of(&s_xbuf[nb ^ 1][tid * 4]),
                        x + base + (size_t)(m + 1) * kN + (size_t)tid * 4);
      } else if (tid >= 256 && tid < 384) {
        const int t = tid - 256;
        async_load_b128(lds_off_of(&s_mbuf[nb ^ 1][t * 4]),
                        xmask + base + (size_t)(m + 1) * kN + (size_t)t * 4);
      }
    }
#else
    xv = x[base + (size_t)m * kN + tid];
    mv = xmask[base + (size_t)m * kN + tid];
#endif

    // ---- stage A: sigmoid + future-mass contribution of this column ----
    const float sig = 1.0f / (1.0f + __expf(-xv));
    const float fut = fmaxf(0.0f, mv - cs);
    s_fut[tid] = fut;
    __syncthreads();

    // ---- stage B: suffix sums of fut ----
#if defined(USE_WMMA)
    // Within-chunk (16-wide) inclusive suffix sums as fut-chunk x triangular-ones,
    // computed by wave 0 (EXEC all-ones) with V_WMMA_F32_16X16X4_F32.
    if (tid < 32) {
      const int laneN = tid & 15;
      const int hi    = tid >> 4;   // 0: K pair {0,1}; 1: K pair {2,3} of each k-step
      #pragma unroll
      for (int T = 0; T < 2; ++T) {
        v8f acc = {0.f, 0.f, 0.f, 0.f, 0.f, 0.f, 0.f, 0.f};
        const int chunk = T * 16 + laneN;        // A-matrix row M = chunk index
        #pragma unroll
        for (int k = 0; k < 4; ++k) {            // K = 16 as 4 chained K=4 WMMAs
          const int kk = 4 * k + 2 * hi;
          v2f a, bm;
          a.x  = s_fut[chunk * 16 + kk];
          a.y  = s_fut[chunk * 16 + kk + 1];
          bm.x = (kk     >= laneN) ? 1.0f : 0.0f;  // U^T[k][n] = (k >= n)
          bm.y = (kk + 1 >= laneN) ? 1.0f : 0.0f;
          acc = __builtin_amdgcn_wmma_f32_16x16x4_f32(
              /*neg_a=*/false, a, /*neg_b=*/false, bm,
              /*c_mod=*/(short)0, acc, /*reuse_a=*/false, /*reuse_b=*/false);
        }
        #pragma unroll
        for (int g = 0; g < 8; ++g) {            // D: lane N = laneN, row M = g + 8*hi
          const int Mo = g + 8 * hi;
          s_suf[(T * 16 + Mo) * 16 + laneN] = acc[g];
        }
      }
      // cross-chunk exclusive suffix carries; chunk total = suffix at column 0
      float carry = 0.0f;
      for (int c2 = tid + 1; c2 < 32; ++c2) carry += s_suf[c2 * 16];
      s_carry[tid] = carry;
    }
#else
    // fallback: full-row inclusive suffix via tree scan (carries stay zero)
    s_suf[tid] = fut;
    __syncthreads();
    for (int off = 1; off < kN; off <<= 1) {
      const float v = s_suf[tid];
      const float a = (tid + off < kN) ? s_suf[tid + off] : 0.0f;
      __syncthreads();
      s_suf[tid] = v + a;
      __syncthreads();
    }
#endif
    __syncthreads();

    // mfm = exclusive suffix of fut = (inclusive suffix at tid) - fut (+ chunk carry)
    const float mfm = s_suf[tid] - fut + s_carry[tid >> 4];
    s_pack[tid] = make_float4(sig, mv, mfm, cs);
    __syncthreads();

    // ---- stage C: the intrinsic serial row_sum scan (one lane) ----
    if (tid == 0) {
      float s = 0.0f;
      #pragma unroll 4
      for (int n = 0; n < kN; ++n) {
        const float4 q = s_pack[n];  // {b, mask, mfm, cs}
        const float lower = q.y * fmaxf(0.0f, 1.0f - s - q.z);
        const float upper = q.y * fminf(1.0f - s, 1.0f - q.w);
        const float p = lower + q.x * (upper - lower);
        s_p[n] = p;
        s += p;
      }
    }
    __syncthreads();

    // ---- stage D: emit row, update column sums ----
    const float p = s_p[tid];
    cs += p;
    out[base + (size_t)m * kN + tid] = p;
  }
}

extern "C" void kernel_launch(void* const* d_in, const int* in_sizes, int n_in,
                              void* d_out, int out_size, void* d_ws, size_t ws_size,
                              hipStream_t stream) {
  const float* x    = (const float*)d_in[0];
  const float* mask = (const float*)d_in[1];
  float*       out  = (float*)d_out;
  const int    B    = in_sizes[0] / (kN * kN);   // 32
  StickBreaking_65953517797987_kernel<<<dim3(B), dim3(kNT), 0, stream>>>(x, mask, out);
}